// Global_pool_block_34256659153378
// MI455X (gfx1250) — compile-verified
//
#include <hip/hip_runtime.h>

typedef float v2f __attribute__((ext_vector_type(2)));
typedef float v8f __attribute__((ext_vector_type(8)));

#define N_BATCH 16
#define C_IN    2048
#define C_OUT   256
#define HW      1024       // 32*32
#define BN_EPS  1e-5f

// ---------------------------------------------------------------------------
// Kernel 0: zero the S/Q accumulators (needed every call: graph replays)
// ---------------------------------------------------------------------------
__global__ void zero_ws_kernel(float* ws, int n) {
    int i = blockIdx.x * blockDim.x + threadIdx.x;
    if (i < n) ws[i] = 0.0f;
}

// ---------------------------------------------------------------------------
// Kernel 1: y = W (256x2048) @ x[n] (2048x1024) via V_WMMA_F32_16X16X4_F32,
// reduced on the fly to S[n][o] = sum_hw y and Q[n][o] = sum_hw y^2.
//
// Block: 512 threads = 16 waves. Each wave handles 16 output channels
// (o0 = wave*16) and a 64-wide hw slab (hw0 = blockIdx.x*64) for batch
// n = blockIdx.y. All 16 waves stream the SAME x tiles -> WGP$/L2 reuse;
// x (134MB) fits in the 192MB L2 so HBM sees ~one pass over x.
// ---------------------------------------------------------------------------
__global__ __launch_bounds__(512) void gemm_stats_kernel(
    const float* __restrict__ x,   // [16][2048][1024]
    const float* __restrict__ W,   // [256][2048]
    float* __restrict__ S,         // [16][256]
    float* __restrict__ Q)         // [16][256]
{
    const int n    = blockIdx.y;
    const int hw0  = blockIdx.x * 64;
    const int wave = threadIdx.x >> 5;
    const int lane = threadIdx.x & 31;
    const int half = lane >> 4;    // 0: lanes 0-15, 1: lanes 16-31
    const int m    = lane & 15;
    const int o0   = wave * 16;

    const float* xn = x + (size_t)n * C_IN * HW;
    const float* Wr = W + (size_t)(o0 + m) * C_IN;   // row of W for this lane

    v8f acc0 = {}, acc1 = {}, acc2 = {}, acc3 = {};

    #pragma unroll 2
    for (int k = 0; k < C_IN; k += 4) {
        // A fragment: 16x4 f32. ISA layout: lanes 0-15 carry K=(0,1),
        // lanes 16-31 carry K=(2,3); M = lane&15. Contiguous float2 load.
        v2f a = *(const v2f*)(Wr + k + 2 * half);

        // B fragment: 4x16 f32. VGPR0: lanes0-15 K=0 / lanes16-31 K=1;
        // VGPR1: lanes0-15 K=2 / lanes16-31 K=3. N = lane&15.
        const float* xb = xn + (size_t)(k + half) * HW + hw0 + m;
        v2f b0, b1, b2, b3;
        b0.x = xb[0];  b0.y = xb[2 * HW];
        b1.x = xb[16]; b1.y = xb[2 * HW + 16];
        b2.x = xb[32]; b2.y = xb[2 * HW + 32];
        b3.x = xb[48]; b3.y = xb[2 * HW + 48];

#if defined(__gfx1250__)
        acc0 = __builtin_amdgcn_wmma_f32_16x16x4_f32(false, a, false, b0,
                                                     (short)0, acc0, false, false);
        acc1 = __builtin_amdgcn_wmma_f32_16x16x4_f32(false, a, false, b1,
                                                     (short)0, acc1, false, false);
        acc2 = __builtin_amdgcn_wmma_f32_16x16x4_f32(false, a, false, b2,
                                                     (short)0, acc2, false, false);
        acc3 = __builtin_amdgcn_wmma_f32_16x16x4_f32(false, a, false, b3,
                                                     (short)0, acc3, false, false);
#else
        (void)a; (void)b0; (void)b1; (void)b2; (void)b3; // host pass only
#endif
    }

    // C/D layout: VGPR r = row (o0 + r) on lanes 0-15, row (o0 + 8 + r) on
    // lanes 16-31; column = lane&15 within each hw sub-tile.
    float ps[8], pq[8];
    #pragma unroll
    for (int r = 0; r < 8; ++r) {
        float v0 = acc0[r], v1 = acc1[r], v2 = acc2[r], v3 = acc3[r];
        ps[r] = v0 + v1 + v2 + v3;
        pq[r] = v0 * v0 + v1 * v1 + v2 * v2 + v3 * v3;
    }

    // Reduce across the 16 lanes of each half (xor masks 1,2,4,8 stay
    // inside a 16-lane group on wave32).
    #pragma unroll
    for (int r = 0; r < 8; ++r) {
        #pragma unroll
        for (int mask = 1; mask < 16; mask <<= 1) {
            ps[r] += __shfl_xor(ps[r], mask, 32);
            pq[r] += __shfl_xor(pq[r], mask, 32);
        }
    }

    if (m == 0) {  // lanes 0 and 16 hold the row totals
        #pragma unroll
        for (int r = 0; r < 8; ++r) {
            int row = o0 + half * 8 + r;
            __hip_atomic_fetch_add(&S[n * C_OUT + row], ps[r],
                                   __ATOMIC_RELAXED, __HIP_MEMORY_SCOPE_AGENT);
            __hip_atomic_fetch_add(&Q[n * C_OUT + row], pq[r],
                                   __ATOMIC_RELAXED, __HIP_MEMORY_SCOPE_AGENT);
        }
    }
}

// ---------------------------------------------------------------------------
// Kernel 2: BN stats per channel, pooled value per (n,o), broadcast 32x32.
// ---------------------------------------------------------------------------
__global__ __launch_bounds__(256) void finalize_kernel(
    const float* __restrict__ S, const float* __restrict__ Q,
    const float* __restrict__ gamma, const float* __restrict__ beta,
    float* __restrict__ out)       // [16][256][1024]
{
    const int o = blockIdx.x;
    __shared__ float pooled[N_BATCH];
    __shared__ float stats[2];     // mean, inv_std

    if (threadIdx.x == 0) {
        float s = 0.0f, q = 0.0f;
        #pragma unroll
        for (int nn = 0; nn < N_BATCH; ++nn) {
            s += S[nn * C_OUT + o];
            q += Q[nn * C_OUT + o];
        }
        const float inv = 1.0f / (float)(N_BATCH * HW);
        float mean = s * inv;
        float var  = q * inv - mean * mean;
        stats[0] = mean;
        stats[1] = rsqrtf(var + BN_EPS);
    }
    __syncthreads();

    if (threadIdx.x < N_BATCH) {
        int nn = threadIdx.x;
        float pn = S[nn * C_OUT + o] * (1.0f / (float)HW);
        pooled[nn] = gamma[o] * (pn - stats[0]) * stats[1] + beta[o];
    }
    __syncthreads();

    for (int idx = threadIdx.x; idx < N_BATCH * HW; idx += blockDim.x) {
        int nn  = idx >> 10;
        int pix = idx & (HW - 1);
        out[(((size_t)nn * C_OUT + o) << 10) + pix] = pooled[nn];
    }
}

// ---------------------------------------------------------------------------
extern "C" void kernel_launch(void* const* d_in, const int* in_sizes, int n_in,
                              void* d_out, int out_size, void* d_ws, size_t ws_size,
                              hipStream_t stream) {
    const float* x     = (const float*)d_in[0];
    const float* W     = (const float*)d_in[1];
    const float* gamma = (const float*)d_in[2];
    const float* beta  = (const float*)d_in[3];
    float* out = (float*)d_out;

    float* S = (float*)d_ws;              // [16][256]
    float* Q = S + N_BATCH * C_OUT;       // [16][256]

    const int acc_n = 2 * N_BATCH * C_OUT;
    zero_ws_kernel<<<(acc_n + 255) / 256, 256, 0, stream>>>(S, acc_n);

    dim3 grid(HW / 64, N_BATCH);          // 16 x 16 blocks, 16 waves each
    gemm_stats_kernel<<<grid, 512, 0, stream>>>(x, W, S, Q);

    finalize_kernel<<<C_OUT, 256, 0, stream>>>(S, Q, gamma, beta, out);
}